// MultiGCNLayers_73735998537947
// MI455X (gfx1250) — compile-verified
//
#include <hip/hip_runtime.h>

// MultiGCNLayers on gfx1250 (MI455X).
// - GEMMs via V_WMMA_F32_16X16X4_F32 (fp32, matches reference precision).
//   B operand staged in LDS k-pair-interleaved so each fragment is one
//   aligned ds_load_b64 directly into an even VGPR pair (no repack movs).
// - Edge aggregation via L2-resident gather + global fp32 atomics
//   (lin/agg are 12.8 MB each << 192 MB L2).
// - LayerNorm via wave32 shfl_xor reductions.

#define GCN_N 50000
#define GCN_E 800000
#define GCN_DIN 128
#define GCN_DH 64
#define GCN_C 4
#define GCN_L 3

typedef __attribute__((ext_vector_type(2))) float v2f;
typedef __attribute__((ext_vector_type(8))) float v8f;

// ---------------------------------------------------------------- utilities
__global__ void zero_f32(float* __restrict__ p, int n) {
  int i = blockIdx.x * blockDim.x + threadIdx.x;
  if (i < n) p[i] = 0.0f;
}

__global__ void deg_count(const int* __restrict__ dst, float* __restrict__ deg, int e) {
  int i = blockIdx.x * blockDim.x + threadIdx.x;
  if (i < e) atomicAdd(&deg[dst[i]], 1.0f);
}

__global__ void deg_to_dinv(float* __restrict__ d, int n) {
  int i = blockIdx.x * blockDim.x + threadIdx.x;
  if (i < n) d[i] = rsqrtf(d[i] + 1.0f);
}

// ---------------------------------------------------------------- WMMA GEMM
// out[N, 64] = H[N, K] @ W[K, 64]
// One wave per 16-row tile; four 16x16 f32 accumulators cover 64 output cols.
// A fragment (16x4 f32): lane L (m = L&15, hi = L>>4) holds
//   H[row0+m][k+2*hi], H[row0+m][k+2*hi+1]  -> one global_load_b64.
// B fragment (4x16 f32): lane L needs {W[kb][col], W[kb+1][col]} with
//   kb = k + 2*hi (even). LDS holds W as k-pair-interleaved float2:
//   sW2[(k>>1)*64 + j] = {W[k][j], W[k+1][j]}  -> one ds_load_b64 per frag.
// C/D (16x16 f32): VGPR r -> row r+8*hi, col = lane&15.
template <int K>
__global__ __launch_bounds__(128) void gemm_wmma(const float* __restrict__ H,
                                                 const float* __restrict__ W,
                                                 float* __restrict__ out,
                                                 int ntiles) {
  __shared__ v2f sW2[(K / 2) * GCN_DH];
  {
    float* sWf = (float*)sW2;
    for (int i = threadIdx.x; i < K * GCN_DH; i += 128) {
      int k = i >> 6;          // row of W
      int j = i & 63;          // col of W
      sWf[(((k >> 1) * GCN_DH) + j) * 2 + (k & 1)] = W[i];
    }
  }
  __syncthreads();

  const int wave = threadIdx.x >> 5;
  const int lane = threadIdx.x & 31;
  const int tile = blockIdx.x * 4 + wave;          // wave-uniform
  if (tile >= ntiles) return;                      // EXEC stays all-ones per wave

  const int row0 = tile * 16;
  const int m  = lane & 15;
  const int hi = lane >> 4;
  const float* arow = H + (size_t)(row0 + m) * K + 2 * hi;
  const v2f* brow = sW2 + (size_t)hi * GCN_DH + m; // pair row hi, col base m

  v8f acc[4] = {};
  for (int k = 0; k < K; k += 4) {
    v2f a = *(const v2f*)(arow + k);               // global_load_b64
    const v2f* bp = brow + (k >> 1) * GCN_DH;      // pair index (k+2*hi)/2
#pragma unroll
    for (int n = 0; n < 4; ++n) {
      v2f b = bp[n * 16];                          // ds_load_b64
      acc[n] = __builtin_amdgcn_wmma_f32_16x16x4_f32(
          /*neg_a=*/false, a, /*neg_b=*/false, b,
          /*c_mod=*/(short)0, acc[n], /*reuse_a=*/false, /*reuse_b=*/false);
    }
  }

#pragma unroll
  for (int n = 0; n < 4; ++n)
#pragma unroll
    for (int r = 0; r < 8; ++r)
      out[(size_t)(row0 + r + 8 * hi) * GCN_DH + n * 16 + m] = acc[n][r];
}

// ---------------------------------------------------------------- scatter
// agg[dst] += lin[src] * dinv[src]*dinv[dst], 32 lanes per edge, float2/lane.
__global__ void scatter_edges(const float* __restrict__ lin,
                              const int* __restrict__ src,
                              const int* __restrict__ dst,
                              const float* __restrict__ dinv,
                              float* __restrict__ agg, int e) {
  int t = blockIdx.x * blockDim.x + threadIdx.x;
  int ed = t >> 5;
  if (ed >= e) return;
  int lane = t & 31;
  int s = src[ed], d = dst[ed];
  float w = dinv[s] * dinv[d];
  const float2 v = *(const float2*)(lin + (size_t)s * GCN_DH + 2 * lane);
  float* ap = agg + (size_t)d * GCN_DH + 2 * lane;
  atomicAdd(ap,     v.x * w);
  atomicAdd(ap + 1, v.y * w);
}

// ---------------------------------------------------------------- combine
// y = agg + dinv^2 * lin + bias;  h = relu(y) + y
__global__ void combine_relu(const float* __restrict__ agg,
                             const float* __restrict__ lin,
                             const float* __restrict__ dinv,
                             const float* __restrict__ bias,
                             float* __restrict__ hout, int n) {
  int i = blockIdx.x * blockDim.x + threadIdx.x;
  if (i >= n) return;
  int node = i >> 6;
  int j = i & 63;
  float dv = dinv[node];
  float y = agg[i] + dv * dv * lin[i] + bias[j];
  hout[i] = fmaxf(y, 0.0f) + y;
}

// ---------------------------------------------------------------- layernorm
// One wave32 per node row (64 floats -> float2 per lane), shfl_xor reduce.
__global__ void layernorm_rows(const float* __restrict__ h,
                               const float* __restrict__ gamma,
                               const float* __restrict__ beta,
                               float* __restrict__ out, int nodes) {
  int t = blockIdx.x * blockDim.x + threadIdx.x;
  int node = t >> 5;
  if (node >= nodes) return;
  int lane = t & 31;
  const float2 v = *(const float2*)(h + (size_t)node * GCN_DH + 2 * lane);
  float s  = v.x + v.y;
  float sq = v.x * v.x + v.y * v.y;
#pragma unroll
  for (int off = 16; off > 0; off >>= 1) {
    s  += __shfl_xor(s,  off, 32);
    sq += __shfl_xor(sq, off, 32);
  }
  float mu  = s * (1.0f / GCN_DH);
  float var = sq * (1.0f / GCN_DH) - mu * mu;
  float rs  = rsqrtf(var + 1e-6f);
  const float2 g = *(const float2*)(gamma + 2 * lane);
  const float2 b = *(const float2*)(beta + 2 * lane);
  float2 o;
  o.x = (v.x - mu) * rs * g.x + b.x;
  o.y = (v.y - mu) * rs * g.y + b.y;
  *(float2*)(out + (size_t)node * GCN_DH + 2 * lane) = o;
}

// ---------------------------------------------------------------- launcher
extern "C" void kernel_launch(void* const* d_in, const int* in_sizes, int n_in,
                              void* d_out, int out_size, void* d_ws, size_t ws_size,
                              hipStream_t stream) {
  (void)in_sizes; (void)n_in; (void)out_size; (void)ws_size;

  const float* x    = (const float*)d_in[0];           // [N, 128]
  const int*   edge = (const int*)d_in[1];             // [2, E]
  const int*   srcI = edge;
  const int*   dstI = edge + GCN_E;
  const float* W0   = (const float*)d_in[3];           // [C, 128, 64]
  const float* b0   = (const float*)d_in[4];           // [C, 64]
  const float* Wl   = (const float*)d_in[5];           // [C, 2, 64, 64]
  const float* bl   = (const float*)d_in[6];           // [C, 2, 64]
  const float* gm   = (const float*)d_in[7];           // [64]
  const float* bt   = (const float*)d_in[8];           // [64]
  float* out = (float*)d_out;                          // [C, N, 64]

  float* ws   = (float*)d_ws;
  float* dinv = ws;                                    // N
  float* lin  = dinv + GCN_N;                          // N*64
  float* agg  = lin + (size_t)GCN_N * GCN_DH;          // N*64
  float* h    = agg + (size_t)GCN_N * GCN_DH;          // N*64

  const int NDH    = GCN_N * GCN_DH;                   // 3.2M
  const int ntiles = GCN_N / 16;                       // 3125 exactly
  const dim3 blk(256);
  const int gN    = (GCN_N + 255) / 256;
  const int gE    = (GCN_E + 255) / 256;
  const int gNDH  = (NDH + 255) / 256;
  const int gScat = (GCN_E * 32) / 256;                // 100000
  const int gLN   = (GCN_N * 32 + 255) / 256;
  const int gGemm = (ntiles + 3) / 4;                  // 4 waves / block

  // degree -> dinv = rsqrt(deg + 1)
  zero_f32<<<gN, blk, 0, stream>>>(dinv, GCN_N);
  deg_count<<<gE, blk, 0, stream>>>(dstI, dinv, GCN_E);
  deg_to_dinv<<<gN, blk, 0, stream>>>(dinv, GCN_N);

  for (int c = 0; c < GCN_C; ++c) {
    // ---- layer 0: K = 128, input = x
    gemm_wmma<GCN_DIN><<<gGemm, dim3(128), 0, stream>>>(
        x, W0 + (size_t)c * GCN_DIN * GCN_DH, lin, ntiles);
    zero_f32<<<gNDH, blk, 0, stream>>>(agg, NDH);
    scatter_edges<<<gScat, blk, 0, stream>>>(lin, srcI, dstI, dinv, agg, GCN_E);
    combine_relu<<<gNDH, blk, 0, stream>>>(agg, lin, dinv,
                                           b0 + (size_t)c * GCN_DH, h, NDH);
    // ---- layers 1..2: K = 64, input = h
    for (int l = 0; l < GCN_L - 1; ++l) {
      gemm_wmma<GCN_DH><<<gGemm, dim3(128), 0, stream>>>(
          h, Wl + ((size_t)c * (GCN_L - 1) + l) * GCN_DH * GCN_DH, lin, ntiles);
      zero_f32<<<gNDH, blk, 0, stream>>>(agg, NDH);
      scatter_edges<<<gScat, blk, 0, stream>>>(lin, srcI, dstI, dinv, agg, GCN_E);
      combine_relu<<<gNDH, blk, 0, stream>>>(agg, lin, dinv,
                                             bl + ((size_t)c * (GCN_L - 1) + l) * GCN_DH,
                                             h, NDH);
    }
    // ---- per-channel LayerNorm into d_out[c]
    layernorm_rows<<<gLN, blk, 0, stream>>>(h, gm, bt,
                                            out + (size_t)c * NDH, GCN_N);
  }
}